// GatDot_rel_27075473834502
// MI455X (gfx1250) — compile-verified
//
#include <hip/hip_runtime.h>
#include <cmath>

#define DD 256
#define BB 256
#define NN 2048

typedef float v2f __attribute__((ext_vector_type(2)));
typedef float v8f __attribute__((ext_vector_type(8)));

// ---------------------------------------------------------------------------
// 256x256x256 fp32 GEMM using V_WMMA_F32_16X16X4_F32.
// One wave (32 threads) computes one 16x16 C tile, looping K in steps of 4.
// MODE 0: C = A @ Bsrc^T   (Bsrc row-major [n][k])  -> used for Q @ W1^T
// MODE 1: C = A @ Bsrc     (Bsrc row-major [k][n])  -> used for q @ W2
// A 16x4 layout : lanes 0-15 -> M=lane, K = {k0,k0+1}; lanes 16-31 -> K = {k0+2,k0+3}
// B 4x16 layout : lanes 0-15 -> N=lane, K = {k0,k0+1}; lanes 16-31 -> K = {k0+2,k0+3}
// C 16x16 layout: VGPR r -> row m0 + r + 8*half, col n0 + (lane&15)
// HAS_BIAS is compile-time; bias value loaded once, latency hidden by k-loop.
// ---------------------------------------------------------------------------
template <int MODE, bool HAS_BIAS>
__global__ void wmma_gemm256(const float* __restrict__ A,
                             const float* __restrict__ Bsrc,
                             const float* __restrict__ bias,
                             float* __restrict__ C) {
  const int m0   = blockIdx.y * 16;
  const int n0   = blockIdx.x * 16;
  const int lane = threadIdx.x;      // 0..31, one wave per block
  const int half = lane >> 4;
  const int idx  = lane & 15;
  const int col  = n0 + idx;

  const float bv = HAS_BIAS ? bias[col] : 0.0f;   // hoisted, single load

  v8f c = {};
  for (int k0 = 0; k0 < DD; k0 += 4) {
    const int kA = k0 + 2 * half;
    v2f a, b;
    a.x = A[(m0 + idx) * DD + kA];
    a.y = A[(m0 + idx) * DD + kA + 1];
    if (MODE == 0) {
      b.x = Bsrc[(n0 + idx) * DD + kA];
      b.y = Bsrc[(n0 + idx) * DD + kA + 1];
    } else {
      b.x = Bsrc[kA * DD + (n0 + idx)];
      b.y = Bsrc[(kA + 1) * DD + (n0 + idx)];
    }
    c = __builtin_amdgcn_wmma_f32_16x16x4_f32(false, a, false, b,
                                              (short)0, c, false, false);
  }

#pragma unroll
  for (int r = 0; r < 8; ++r) {
    const int row = m0 + r + 8 * half;
    C[row * DD + col] = c[r] + bv;
  }
}

// qb2[b] = q[b,:] . b2
__global__ void dot_qb2(const float* __restrict__ q,
                        const float* __restrict__ b2,
                        float* __restrict__ qb2) {
  __shared__ float red[256];
  const int b = blockIdx.x, t = threadIdx.x;
  red[t] = q[b * DD + t] * b2[t];
  __syncthreads();
  for (int s = 128; s > 0; s >>= 1) {
    if (t < s) red[t] += red[t + s];
    __syncthreads();
  }
  if (t == 0) qb2[b] = red[0];
}

// y[0] = W3 . rel_emb[0] + b3 ; y[1] = W3 . rel_emb[1] + b3
__global__ void rel_bias(const float* __restrict__ W3,
                         const float* __restrict__ b3,
                         const float* __restrict__ rel_emb,
                         float* __restrict__ y) {
  __shared__ float r0[256];
  __shared__ float r1[256];
  const int t = threadIdx.x;
  const float w = W3[t];
  r0[t] = w * rel_emb[t];
  r1[t] = w * rel_emb[DD + t];
  __syncthreads();
  for (int s = 128; s > 0; s >>= 1) {
    if (t < s) { r0[t] += r0[t + s]; r1[t] += r1[t + s]; }
    __syncthreads();
  }
  if (t == 0) { y[0] = r0[0] + b3[0]; y[1] = r1[0] + b3[0]; }
}

// alpha[b,n] = K[b,n,:] . qp[b,:] + qb2[b] + y[s_mask] - (1-adj)*1e30
// One wave per row; lane l covers d = l*8 .. l*8+7 (two coalesced float4 loads).
__global__ void alpha_kernel(const float* __restrict__ Kmat,
                             const float* __restrict__ qp,
                             const float* __restrict__ qb2,
                             const float* __restrict__ y,
                             const float* __restrict__ adj,
                             const int* __restrict__ s_mask,
                             float* __restrict__ alpha) {
  const int ROWS = 64;                       // rows per block (8 waves x 8)
  const int b    = blockIdx.x / (NN / ROWS);
  const int tile = blockIdx.x % (NN / ROWS);
  const int lane = threadIdx.x & 31;
  const int wave = threadIdx.x >> 5;

  const float4* qp4 = (const float4*)(qp + (size_t)b * DD + lane * 8);
  const float4 p0 = qp4[0], p1 = qp4[1];
  const float base = qb2[b];
  const float y0 = y[0], y1 = y[1];

  for (int r = 0; r < 8; ++r) {
    const int n = tile * ROWS + wave * 8 + r;
    const float4* kr =
        (const float4*)(Kmat + ((size_t)b * NN + n) * DD + lane * 8);
    const float4 k0 = kr[0], k1 = kr[1];
    float s = k0.x * p0.x + k0.y * p0.y + k0.z * p0.z + k0.w * p0.w +
              k1.x * p1.x + k1.y * p1.y + k1.z * p1.z + k1.w * p1.w;
    for (int off = 16; off > 0; off >>= 1) s += __shfl_down(s, off);
    if (lane == 0) {
      const size_t i = (size_t)b * NN + n;
      const float yv = s_mask[i] ? y1 : y0;
      alpha[i] = s + base + yv - (1.0f - adj[i]) * 1e30f;
    }
  }
}

// In-place stable softmax over N per batch row.
__global__ void softmax_kernel(float* __restrict__ w) {
  __shared__ float red[256];
  const int b = blockIdx.x, t = threadIdx.x;
  float* row = w + (size_t)b * NN;

  float vals[8];
  float m = -INFINITY;
#pragma unroll
  for (int i = 0; i < 8; ++i) {
    vals[i] = row[t + i * 256];
    m = fmaxf(m, vals[i]);
  }
  red[t] = m;
  __syncthreads();
  for (int s = 128; s > 0; s >>= 1) {
    if (t < s) red[t] = fmaxf(red[t], red[t + s]);
    __syncthreads();
  }
  m = red[0];
  __syncthreads();

  float sum = 0.f;
#pragma unroll
  for (int i = 0; i < 8; ++i) {
    vals[i] = expf(vals[i] - m);
    sum += vals[i];
  }
  red[t] = sum;
  __syncthreads();
  for (int s = 128; s > 0; s >>= 1) {
    if (t < s) red[t] += red[t + s];
    __syncthreads();
  }
  const float inv = 1.0f / red[0];
#pragma unroll
  for (int i = 0; i < 8; ++i) row[t + i * 256] = vals[i] * inv;
}

__global__ void zero_kernel(float* __restrict__ p, int n) {
  const int i = blockIdx.x * 256 + threadIdx.x;
  if (i < n) p[i] = 0.f;
}

// attn_sum[b,:] += sum_{n in segment} w[b,n] * V[b,n,:]
// thread t owns column d = t (coalesced 1KB line per n). 8 segments per batch.
__global__ void attnsum_kernel(const float* __restrict__ w,
                               const float* __restrict__ V,
                               float* __restrict__ out) {
  const int SEG = NN / 8;                    // 256 rows per segment
  const int b   = blockIdx.x >> 3;
  const int seg = blockIdx.x & 7;
  const int t   = threadIdx.x;

  __shared__ float lw[SEG];
  lw[t] = w[(size_t)b * NN + seg * SEG + t];
  __syncthreads();

  float acc = 0.f;
  const float* vp = V + ((size_t)b * NN + (size_t)seg * SEG) * DD + t;
#pragma unroll 8
  for (int i = 0; i < SEG; ++i) acc += lw[i] * vp[(size_t)i * DD];
  atomicAdd(&out[b * DD + t], acc);
}

extern "C" void kernel_launch(void* const* d_in, const int* in_sizes, int n_in,
                              void* d_out, int out_size, void* d_ws,
                              size_t ws_size, hipStream_t stream) {
  const float* Q       = (const float*)d_in[0];
  const float* Kmat    = (const float*)d_in[1];
  const float* V       = (const float*)d_in[2];
  const float* adj     = (const float*)d_in[3];
  const int*   s_mask  = (const int*)d_in[4];
  const float* W1      = (const float*)d_in[5];
  const float* b1      = (const float*)d_in[6];
  const float* W2      = (const float*)d_in[7];
  const float* b2      = (const float*)d_in[8];
  const float* W3      = (const float*)d_in[9];
  const float* b3      = (const float*)d_in[10];
  const float* rel_emb = (const float*)d_in[11];

  float* out      = (float*)d_out;           // [0, B*N): attn_weight; then attn_sum
  float* attn_w   = out;
  float* attn_sum = out + (size_t)BB * NN;

  float* ws_q   = (float*)d_ws;              // B*D
  float* ws_qp  = ws_q + BB * DD;            // B*D
  float* ws_qb2 = ws_qp + BB * DD;           // B
  float* ws_y   = ws_qb2 + BB;               // 2

  // q = Q @ W1^T + b1      (WMMA fp32)
  wmma_gemm256<0, true><<<dim3(16, 16), 32, 0, stream>>>(Q, W1, b1, ws_q);
  // qp = q @ W2            (WMMA fp32)
  wmma_gemm256<1, false><<<dim3(16, 16), 32, 0, stream>>>(ws_q, W2, nullptr,
                                                          ws_qp);
  // qb2[b] = q[b,:] . b2
  dot_qb2<<<BB, 256, 0, stream>>>(ws_q, b2, ws_qb2);
  // y0/y1 relation bias scalars
  rel_bias<<<1, 256, 0, stream>>>(W3, b3, rel_emb, ws_y);
  // masked scores -> attn_weight region (raw)
  alpha_kernel<<<BB * (NN / 64), 256, 0, stream>>>(Kmat, ws_qp, ws_qb2, ws_y,
                                                   adj, s_mask, attn_w);
  // in-place softmax
  softmax_kernel<<<BB, 256, 0, stream>>>(attn_w);
  // attn_sum = w @ V
  zero_kernel<<<(BB * DD + 255) / 256, 256, 0, stream>>>(attn_sum, BB * DD);
  attnsum_kernel<<<BB * 8, 256, 0, stream>>>(attn_w, V, attn_sum);
}